// MinkowskiInstanceNorm_88656714925515
// MI455X (gfx1250) — compile-verified
//
#include <hip/hip_runtime.h>
#include <hip/hip_bf16.h>

// MinkowskiInstanceNorm for MI455X (gfx1250, wave32).
// N=1e6 rows, C=128 channels, B=16 sorted segments, f32.
// Two-pass, HBM-bound (~1.5GB traffic => ~66us floor at 23.3 TB/s).
// Pass-1 segment reduction uses V_WMMA_F32_16X16X4_F32 with an all-ones
// A matrix as a tensor-core column-sum engine (64 accumulates per WMMA,
// co-executes with loads). Segment boundaries are precomputed so the hot
// loop is branch-free and carries no accumulator phis.

#define C_CH   128
#define B_SEG  16
#define EPSV   1e-5f
#define CHUNK_ROWS 512

typedef __attribute__((ext_vector_type(2))) float v2f;
typedef __attribute__((ext_vector_type(8))) float v8f;

// ---------------- zero stats ----------------
__global__ void min_zero_kernel(float* p, int n) {
    int i = blockIdx.x * blockDim.x + threadIdx.x;
    if (i < n) p[i] = 0.0f;
}

// ---------------- segment boundaries via binary search ----------------
__global__ void min_bounds_kernel(const int* __restrict__ idx, int* __restrict__ bnd, int n) {
    int t = threadIdx.x;
    if (t > B_SEG) return;
    if (t == B_SEG) { bnd[t] = n; return; }
    // first i with idx[i] >= t   (idx sorted ascending)
    int lo = 0, hi = n;
    while (lo < hi) {
        int mid = (lo + hi) >> 1;
        if (idx[mid] < t) lo = mid + 1; else hi = mid;
    }
    bnd[t] = lo;
}

// ---------------- pass 1: per-segment sum / sumsq via WMMA ones-reduction --
// Block = 256 threads = 8 waves. Wave w owns channels [16w, 16w+16).
// Each block handles CHUNK_ROWS contiguous rows, split into segment-uniform
// ranges via the precomputed boundaries. Per 4-row group the wave loads a
// 4x16 f32 B tile and does D = ones(16x4) x B + D: every row of D holds the
// running per-channel sums (element 0 of the accumulator on lanes 0-15).
__global__ void min_stats_kernel(const float* __restrict__ x,
                                 const int*   __restrict__ bnd,
                                 float* __restrict__ gsum,
                                 float* __restrict__ gsq,
                                 int n) {
    const int lane     = threadIdx.x & 31;
    const int wave     = threadIdx.x >> 5;
    const int chanBase = wave * 16;
    const int nlo      = lane & 15;   // channel within slice / N index
    const int khalf    = lane >> 4;   // 0: rows r,r+1   1: rows r+2,r+3

    const int rowStart = blockIdx.x * CHUNK_ROWS;
    const int rowEnd   = (rowStart + CHUNK_ROWS < n) ? (rowStart + CHUNK_ROWS) : n;

    const v8f vzero = {0.f,0.f,0.f,0.f,0.f,0.f,0.f,0.f};
    const v2f onesA = {1.0f, 1.0f};

    for (int s = 0; s < B_SEG; ++s) {
        int r0 = bnd[s];     if (r0 < rowStart) r0 = rowStart;
        int r1 = bnd[s + 1]; if (r1 > rowEnd)   r1 = rowEnd;
        if (r0 >= r1) continue;                 // wave-uniform (block-uniform)

        v8f accS = vzero;
        v8f accQ = vzero;
        int r = r0;
        const float* p = x + (size_t)(r + 2 * khalf) * C_CH + chanBase + nlo;

        // unrolled x2: 8 rows per iteration, 4 clause-grouped loads, 4 WMMAs
        for (; r + 8 <= r1; r += 8) {
            float a0 = p[0];
            float a1 = p[C_CH];
            float b0 = p[4 * C_CH];
            float b1 = p[5 * C_CH];
            __builtin_prefetch(p + 8 * C_CH, 0, 1);   // global_prefetch_b8
            v2f Av = { a0, a1 };
            v2f Aq = { a0 * a0, a1 * a1 };
            v2f Bv = { b0, b1 };
            v2f Bq = { b0 * b0, b1 * b1 };
            accS = __builtin_amdgcn_wmma_f32_16x16x4_f32(
                false, onesA, false, Av, (short)0, accS, false, false);
            accQ = __builtin_amdgcn_wmma_f32_16x16x4_f32(
                false, onesA, false, Aq, (short)0, accQ, false, false);
            accS = __builtin_amdgcn_wmma_f32_16x16x4_f32(
                false, onesA, false, Bv, (short)0, accS, false, false);
            accQ = __builtin_amdgcn_wmma_f32_16x16x4_f32(
                false, onesA, false, Bq, (short)0, accQ, false, false);
            p += 8 * C_CH;
        }
        // possibly one remaining full 4-row group
        if (r + 4 <= r1) {
            float a0 = p[0];
            float a1 = p[C_CH];
            v2f Av = { a0, a1 };
            v2f Aq = { a0 * a0, a1 * a1 };
            accS = __builtin_amdgcn_wmma_f32_16x16x4_f32(
                false, onesA, false, Av, (short)0, accS, false, false);
            accQ = __builtin_amdgcn_wmma_f32_16x16x4_f32(
                false, onesA, false, Aq, (short)0, accQ, false, false);
            r += 4;
        }
        // scalar tail rows [r, r1): lanes 0-15 only (EXEC restored before
        // the next segment's WMMAs)
        float tS = 0.0f, tQ = 0.0f;
        if (khalf == 0) {
            for (int rr = r; rr < r1; ++rr) {
                float v = x[(size_t)rr * C_CH + chanBase + nlo];
                tS += v;
                tQ += v * v;
            }
        }
        // one flush per (block x segment): lane L<16 holds D[M=0][N=L]
        if (lane < 16) {
            atomicAdd(&gsum[s * C_CH + chanBase + nlo], accS[0] + tS);
            atomicAdd(&gsq [s * C_CH + chanBase + nlo], accQ[0] + tQ);
        }
    }
}

// ---------------- finalize mean / instd ----------------
__global__ void min_finalize_kernel(const float* __restrict__ gsum,
                                    const float* __restrict__ gsq,
                                    const int*   __restrict__ bnd,
                                    float* __restrict__ mean,
                                    float* __restrict__ instd) {
    int i = blockIdx.x * blockDim.x + threadIdx.x;
    if (i >= B_SEG * C_CH) return;
    int b = i / C_CH;
    int cnt_i = bnd[b + 1] - bnd[b];
    float cnt = (float)(cnt_i < 1 ? 1 : cnt_i);
    float m = gsum[i] / cnt;
    float v = gsq[i] / cnt - m * m;
    v = v < 0.0f ? 0.0f : v;
    mean[i]  = m;
    instd[i] = rsqrtf(v + EPSV);
}

// ---------------- pass 2: vectorized normalize (b128 loads/stores) --------
__global__ void min_norm_kernel(const float* __restrict__ x,
                                const int*   __restrict__ idx,
                                const float* __restrict__ w,
                                const float* __restrict__ bias,
                                const float* __restrict__ mean,
                                const float* __restrict__ instd,
                                float* __restrict__ out,
                                int nvec) {
    int i = blockIdx.x * blockDim.x + threadIdx.x;     // float4 index
    if (i >= nvec) return;
    int row = i >> 5;                                  // C/4 = 32 vec4 per row
    int c4  = i & 31;
    int seg = idx[row];
    int sc  = seg * 32 + c4;

    float4 xv = ((const float4*)x)[i];
    float4 m  = ((const float4*)mean)[sc];
    float4 is = ((const float4*)instd)[sc];
    float4 wv = ((const float4*)w)[c4];
    float4 bv = ((const float4*)bias)[c4];

    float4 o;
    o.x = (xv.x - m.x) * is.x * wv.x + bv.x;
    o.y = (xv.y - m.y) * is.y * wv.y + bv.y;
    o.z = (xv.z - m.z) * is.z * wv.z + bv.z;
    o.w = (xv.w - m.w) * is.w * wv.w + bv.w;
    ((float4*)out)[i] = o;
}

extern "C" void kernel_launch(void* const* d_in, const int* in_sizes, int n_in,
                              void* d_out, int out_size, void* d_ws, size_t ws_size,
                              hipStream_t stream) {
    const float* x    = (const float*)d_in[0];
    const int*   idx  = (const int*)  d_in[1];
    const float* w    = (const float*)d_in[2];
    const float* bias = (const float*)d_in[3];
    float*       out  = (float*)d_out;
    const int n = in_sizes[1];                 // N rows (batch_idx count)

    // workspace layout (floats)
    float* gsum  = (float*)d_ws;               // B*C = 2048
    float* gsq   = gsum  + B_SEG * C_CH;       // 2048
    float* mean  = gsq   + B_SEG * C_CH;       // 2048
    float* instd = mean  + B_SEG * C_CH;       // 2048
    int*   bnd   = (int*)(instd + B_SEG * C_CH); // 17 ints

    min_zero_kernel<<<(2 * B_SEG * C_CH + 255) / 256, 256, 0, stream>>>(gsum, 2 * B_SEG * C_CH);
    min_bounds_kernel<<<1, 32, 0, stream>>>(idx, bnd, n);

    int nblocks = (n + CHUNK_ROWS - 1) / CHUNK_ROWS;
    min_stats_kernel<<<nblocks, 256, 0, stream>>>(x, bnd, gsum, gsq, n);

    min_finalize_kernel<<<(B_SEG * C_CH + 255) / 256, 256, 0, stream>>>(gsum, gsq, bnd, mean, instd);

    int nvec = n * (C_CH / 4);
    min_norm_kernel<<<(nvec + 255) / 256, 256, 0, stream>>>(x, idx, w, bias, mean, instd, out, nvec);
}